// WaveFunction_7825430413845
// MI455X (gfx1250) — compile-verified
//
#include <hip/hip_runtime.h>

// MPS contraction for MI455X (gfx1250), wave32 + WMMA fp32.
//
// reference: vec0 = ones[B,H]; for s in 0..63: vec = vec @ T[s, x[b,s]]; out = vec.sum(-1)
//   B=2048, SITES=64, PHYS=2, H=128, fp32 everywhere.
//
// Tensorization: 16-sample tiles. vec_new = mask0(vec) @ T[s,0] + mask1(vec) @ T[s,1]
// computed as V_WMMA_F32_16X16X4_F32 accumulations (fp32, bit-faithful precision path).

typedef float v2f __attribute__((ext_vector_type(2)));
typedef float v8f __attribute__((ext_vector_type(8)));

#define SITES   64
#define PHYS    2
#define HID     128
#define BATCH   2048
#define MTILE   16            // batch samples per workgroup (WMMA M)
#define NWAVES  8             // 8 wave32 = 256 threads; wave w owns N cols [16w,16w+16)
#define VSTRIDE (HID + 4)     // LDS row pad: A-operand reads become bank-conflict-free

__global__ __launch_bounds__(256, 4)
void mps_wmma_kernel(const int* __restrict__ x,      // [BATCH][SITES]
                     const float* __restrict__ T,    // [SITES][PHYS][HID][HID]
                     float* __restrict__ out)        // [BATCH]
{
    __shared__ float vecbuf[2][MTILE][VSTRIDE];   // double-buffered state vectors (~16.5KB)
    __shared__ int   xs[MTILE][SITES];            // site indices for this batch tile (4KB)

    const int tid  = threadIdx.x;
    const int lane = tid & 31;
    const int wave = tid >> 5;
    const int hl   = lane >> 4;      // half-wave select (K split per ISA A/B layout)
    const int l16  = lane & 15;      // M index (A/C) or N index (B/C) within tile
    const int b0   = blockIdx.x * MTILE;
    const int n0   = wave * 16;      // this wave's output-column base

    // Stage x tile into LDS once (64 sequential site reads come from LDS afterwards).
    for (int i = tid; i < MTILE * SITES; i += 256) {
        const int m = i >> 6, s = i & 63;
        xs[m][s] = x[(b0 + m) * SITES + s];
    }
    // vec0 = ones
    for (int i = tid; i < MTILE * VSTRIDE; i += 256)
        (&vecbuf[0][0][0])[i] = 1.0f;
    __syncthreads();

    int cur = 0;
    for (int s = 0; s < SITES; ++s) {
        const float* __restrict__ T0 = T + (size_t)(s * PHYS + 0) * HID * HID;
        const float* __restrict__ T1 = T0 + (size_t)HID * HID;

        // Prefetch the next site's two matrices (128KB) toward L2/WGP$ while we compute.
        if (s + 1 < SITES) {
            const char* nxt = (const char*)(T + (size_t)((s + 1) * PHYS) * HID * HID);
            #pragma unroll
            for (int j = 0; j < 4; ++j)
                __builtin_prefetch(nxt + (size_t)(tid + 256 * j) * 128, 0, 1);
        }

        const int   xm  = xs[l16][s];                 // this lane's sample site-index
        const float* __restrict__ vin = &vecbuf[cur][0][0];
        float* __restrict__ vout      = &vecbuf[cur ^ 1][0][0];

        v8f c = {};  // 16x16 fp32 accumulator tile (8 VGPRs across wave32)

        #pragma unroll 4
        for (int kt = 0; kt < HID / 4; ++kt) {
            const int k0 = kt * 4;
            const int ka = k0 + 2 * hl;               // ISA A layout: VGPR0=K{0,2}, VGPR1=K{1,3}

            // A operand from LDS: row M=l16, two consecutive K values.
            const float a0 = vin[l16 * VSTRIDE + ka];
            const float a1 = vin[l16 * VSTRIDE + ka + 1];
            v2f aP0, aP1;
            aP0.x = (xm == 0) ? a0 : 0.0f;            // mask -> physical index 0 path
            aP0.y = (xm == 0) ? a1 : 0.0f;
            aP1.x = (xm == 0) ? 0.0f : a0;            // mask -> physical index 1 path
            aP1.y = (xm == 0) ? 0.0f : a1;

            // B operands from global (L2-resident): rows K=ka,ka+1, col N=n0+l16.
            v2f b0v, b1v;
            b0v.x = T0[(ka    ) * HID + n0 + l16];
            b0v.y = T0[(ka + 1) * HID + n0 + l16];
            b1v.x = T1[(ka    ) * HID + n0 + l16];
            b1v.y = T1[(ka + 1) * HID + n0 + l16];

            // D = A1*B1 + (A0*B0 + C): two chained fp32 WMMAs per K-tile.
            c = __builtin_amdgcn_wmma_f32_16x16x4_f32(false, aP0, false, b0v,
                                                      (short)0, c, false, false);
            c = __builtin_amdgcn_wmma_f32_16x16x4_f32(false, aP1, false, b1v,
                                                      (short)0, c, false, false);
        }

        // Scatter C tile to the other LDS buffer. ISA C/D layout:
        // VGPR r, lanes0-15 -> row r, lanes16-31 -> row r+8; col = lane&15.
        #pragma unroll
        for (int r = 0; r < 8; ++r)
            vout[(r + 8 * hl) * VSTRIDE + n0 + l16] = c[r];

        __syncthreads();   // single barrier per site is sufficient with double buffering
        cur ^= 1;
    }

    // Final contraction with ones vector == row sums of the last state.
    if (tid < MTILE) {
        float sum = 0.0f;
        #pragma unroll 8
        for (int n = 0; n < HID; ++n)
            sum += vecbuf[cur][tid][n];
        out[b0 + tid] = sum;
    }
}

extern "C" void kernel_launch(void* const* d_in, const int* in_sizes, int n_in,
                              void* d_out, int out_size, void* d_ws, size_t ws_size,
                              hipStream_t stream) {
    const int*   x = (const int*)d_in[0];     // setup_inputs order: x, tensor
    const float* T = (const float*)d_in[1];
    float*       o = (float*)d_out;           // [BATCH] fp32

    dim3 grid(BATCH / MTILE);   // 128 workgroups
    dim3 block(256);            // 8 wave32
    mps_wmma_kernel<<<grid, block, 0, stream>>>(x, T, o);
}